// SubGraph_23390391894920
// MI455X (gfx1250) — compile-verified
//
#include <hip/hip_runtime.h>
#include <math.h>

// ---------------------------------------------------------------------------
// CDNA5 (gfx1250) implementation of the 3-layer GraphConv-MLP + scatter-max
// + cluster max-pool + column-L2-normalize reference.
//
// Per-node GEMMs run on the matrix units via V_WMMA_F32_16X16X4_F32 (full
// fp32, matching the fp32 reference). Scatter-max uses native B32 atomics
// (monotone-bits int-max / uint-min trick => global_atomic_max/min_b32,
// no CAS loop). Empty segments reproduce the reference's -inf -> 0 fill.
// ---------------------------------------------------------------------------

typedef float v2f __attribute__((ext_vector_type(2)));
typedef float v8f __attribute__((ext_vector_type(8)));

#define NEG_INF (-__builtin_inff())

__device__ __forceinline__ v8f wmma4(v2f a, v2f b, v8f c) {
  // D = A(16x4) * B(4x16) + C(16x16), fp32. 8-arg form:
  // (neg_a, A, neg_b, B, c_mod, C, reuse_a, reuse_b)
  return __builtin_amdgcn_wmma_f32_16x16x4_f32(false, a, false, b,
                                               (short)0, c, false, false);
}

// Order-correct float atomic max built from native integer atomics.
// IEEE-754: for x >= 0 int bits compare like floats; for x < 0 uint bits
// compare in reverse float order. Works with -inf (0xFF800000) init.
__device__ __forceinline__ void atomic_max_f32(float* addr, float v) {
  if (!(v < 0.0f))
    atomicMax((int*)addr, __float_as_int(v));
  else
    atomicMin((unsigned int*)addr, __float_as_uint(v));
}

// ---------------------------------------------------------------------------
// Per-node MLP: h = relu(LN(x@w1 + b1)) @ w2 + b2
// One wave per 16-node tile. Writes h into cols [0,cin) of xout and -inf
// into the aggregation half cols [cin, 2*cin).
// ---------------------------------------------------------------------------
__global__ __launch_bounds__(128) void glp_mlp_kernel(
    const float* __restrict__ xin, const float* __restrict__ w1,
    const float* __restrict__ b1, const float* __restrict__ gam,
    const float* __restrict__ bet, const float* __restrict__ w2,
    const float* __restrict__ b2, float* __restrict__ xout,
    int cin, int ntiles) {
  __shared__ float lds_h[4][16 * 64];  // per-wave 16x64 staging of relu(LN(h))
  const int lane = threadIdx.x & 31;
  const int wave = threadIdx.x >> 5;
  const int tile = blockIdx.x * 4 + wave;
  const int m   = lane & 15;         // row index (A/C lower half) / N column
  const int ko  = (lane >> 4) << 1;  // K offset for upper lane half (A,B)
  const int hi8 = (lane >> 4) << 3;  // +8 row offset for C upper lane half
  const int ldo = cin * 2;
  const bool active = (tile < ntiles);
  const int row0 = tile << 4;
  float* hl = lds_h[wave];

  if (active) {
    // ---- GEMM1: (16 x cin) @ w1(cin x 64), 4 output column tiles ----
    v8f acc[4];
    v8f vz = {};
#pragma unroll
    for (int nt = 0; nt < 4; ++nt) acc[nt] = vz;

    for (int k0 = 0; k0 < cin; k0 += 4) {
      v2f a;
      const float* xr = xin + (size_t)(row0 + m) * cin + k0 + ko;
      a.x = xr[0];
      a.y = xr[1];
#pragma unroll
      for (int nt = 0; nt < 4; ++nt) {
        v2f b;
        b.x = w1[(k0 + ko) * 64 + nt * 16 + m];
        b.y = w1[(k0 + ko + 1) * 64 + nt * 16 + m];
        acc[nt] = wmma4(a, b, acc[nt]);
      }
    }

    // ---- + b1, LayerNorm over the 64 hidden values of each row, ReLU ----
    float mean[8], rstd[8];
#pragma unroll
    for (int j = 0; j < 8; ++j) {
      float s = 0.0f, s2 = 0.0f;
#pragma unroll
      for (int nt = 0; nt < 4; ++nt) {
        float v = acc[nt][j] + b1[nt * 16 + m];
        acc[nt][j] = v;
        s += v;
        s2 += v * v;
      }
      // reduce within each 16-lane half (rows differ between halves)
#pragma unroll
      for (int msk = 1; msk < 16; msk <<= 1) {
        s  += __shfl_xor(s, msk, 32);
        s2 += __shfl_xor(s2, msk, 32);
      }
      float mu  = s * (1.0f / 64.0f);
      float var = s2 * (1.0f / 64.0f) - mu * mu;
      mean[j] = mu;
      rstd[j] = rsqrtf(var + 1e-5f);
    }
#pragma unroll
    for (int j = 0; j < 8; ++j) {
      int r = j + hi8;
#pragma unroll
      for (int nt = 0; nt < 4; ++nt) {
        int col = nt * 16 + m;
        float v = (acc[nt][j] - mean[j]) * rstd[j] * gam[col] + bet[col];
        hl[r * 64 + col] = fmaxf(v, 0.0f);
      }
    }
  }
  __syncthreads();  // also orders the per-wave LDS write->read (DS in-order)

  if (active) {
    // ---- GEMM2: (16 x 64) @ w2(64 x cout), cout == cin ----
    const int nto = cin >> 4;
    for (int nt = 0; nt < nto; ++nt) {
      v8f a2 = {};
#pragma unroll
      for (int k0 = 0; k0 < 64; k0 += 4) {
        v2f a, b;
        a.x = hl[m * 64 + k0 + ko];
        a.y = hl[m * 64 + k0 + ko + 1];
        b.x = w2[(k0 + ko) * cin + nt * 16 + m];
        b.y = w2[(k0 + ko + 1) * cin + nt * 16 + m];
        a2 = wmma4(a, b, a2);
      }
#pragma unroll
      for (int j = 0; j < 8; ++j) {
        int r = j + hi8;
        int col = nt * 16 + m;
        xout[(size_t)(row0 + r) * ldo + col] = a2[j] + b2[col];
      }
    }
    // init aggregation half to -inf for the scatter-max
    for (int r = 0; r < 16; ++r)
      for (int c = lane; c < cin; c += 32)
        xout[(size_t)(row0 + r) * ldo + cin + c] = NEG_INF;
  }
}

// ---------------------------------------------------------------------------
// Edge scatter-max: agg[dst] = max(agg[dst], h[src]) per channel.
// One thread per (edge, 4 channels); float4 gather + 4 native atomics.
// ---------------------------------------------------------------------------
__global__ __launch_bounds__(256) void scatter_max_kernel(
    float* xbuf, const long long* __restrict__ src,
    const long long* __restrict__ dst, int E, int C, int c4sh) {
  long long tid = (long long)blockIdx.x * 256 + threadIdx.x;
  long long total = ((long long)E) << c4sh;
  if (tid >= total) return;
  int e = (int)(tid >> c4sh);
  int c = (int)(tid & ((1 << c4sh) - 1)) << 2;
  int ldo = C * 2;
  int s = (int)src[e];
  int d = (int)dst[e];
  const float4 v = *(const float4*)(xbuf + (size_t)s * ldo + c);
  float* p = xbuf + (size_t)d * ldo + C + c;
  atomic_max_f32(p + 0, v.x);
  atomic_max_f32(p + 1, v.y);
  atomic_max_f32(p + 2, v.z);
  atomic_max_f32(p + 3, v.w);
}

__global__ void fixup_neginf_kernel(float* p, long long n) {
  long long i = (long long)blockIdx.x * blockDim.x + threadIdx.x;
  if (i < n && p[i] == NEG_INF) p[i] = 0.0f;
}

__global__ void init_neginf_kernel(float* p, long long n) {
  long long i = (long long)blockIdx.x * blockDim.x + threadIdx.x;
  if (i < n) p[i] = NEG_INF;
}

// Cluster max-pool over 512-wide final features.
__global__ __launch_bounds__(256) void pool_max_kernel(
    const float* __restrict__ x3, const long long* __restrict__ cluster,
    float* pooled, int N) {
  long long tid = (long long)blockIdx.x * 256 + threadIdx.x;
  long long total = (long long)N * 128;  // 512 / 4
  if (tid >= total) return;
  int n = (int)(tid >> 7);
  int c = (int)(tid & 127) << 2;
  int cl = (int)cluster[n];
  const float4 v = *(const float4*)(x3 + (size_t)n * 512 + c);
  float* p = pooled + (size_t)cl * 512 + c;
  atomic_max_f32(p + 0, v.x);
  atomic_max_f32(p + 1, v.y);
  atomic_max_f32(p + 2, v.z);
  atomic_max_f32(p + 3, v.w);
}

// Per-column L2 norm over the pooled matrix (rows x 512), one block/column.
__global__ __launch_bounds__(256) void col_norm_kernel(
    const float* __restrict__ pooled, float* __restrict__ norms, int rows) {
  __shared__ float red[256];
  int col = blockIdx.x;
  float s = 0.0f;
  for (int r = threadIdx.x; r < rows; r += 256) {
    float v = pooled[(size_t)r * 512 + col];
    s += v * v;
  }
  red[threadIdx.x] = s;
  __syncthreads();
  for (int off = 128; off > 0; off >>= 1) {
    if (threadIdx.x < off) red[threadIdx.x] += red[threadIdx.x + off];
    __syncthreads();
  }
  if (threadIdx.x == 0) norms[col] = sqrtf(red[0]);
}

__global__ void normalize_kernel(const float* __restrict__ pooled,
                                 const float* __restrict__ norms,
                                 float* __restrict__ out, long long n) {
  long long i = (long long)blockIdx.x * blockDim.x + threadIdx.x;
  if (i < n) out[i] = pooled[i] / norms[i & 511];
}

// ---------------------------------------------------------------------------
extern "C" void kernel_launch(void* const* d_in, const int* in_sizes, int n_in,
                              void* d_out, int out_size, void* d_ws,
                              size_t ws_size, hipStream_t stream) {
  const float* x0 = (const float*)d_in[0];
  const long long* eidx = (const long long*)d_in[1];      // int64 (2,E)
  const long long* cluster = (const long long*)d_in[2];   // int64 (N,)
  const int N = in_sizes[0] / 64;
  const int E = in_sizes[1] / 2;
  const int NC = 5000;  // n_clusters (fixed by reference setup)
  const long long* src = eidx;
  const long long* dst = eidx + E;

  const float* W1[3] = {(const float*)d_in[5],  (const float*)d_in[11], (const float*)d_in[17]};
  const float* B1[3] = {(const float*)d_in[6],  (const float*)d_in[12], (const float*)d_in[18]};
  const float* G[3]  = {(const float*)d_in[7],  (const float*)d_in[13], (const float*)d_in[19]};
  const float* BE[3] = {(const float*)d_in[8],  (const float*)d_in[14], (const float*)d_in[20]};
  const float* W2[3] = {(const float*)d_in[9],  (const float*)d_in[15], (const float*)d_in[21]};
  const float* B2[3] = {(const float*)d_in[10], (const float*)d_in[16], (const float*)d_in[22]};

  // Workspace layout (floats): x1(N*128) | x2(N*256) | x3(N*512) |
  // pooled(NC*512) | norms(512)   -> ~190 MB total for the given shapes.
  float* ws = (float*)d_ws;
  float* x1 = ws;
  float* x2 = x1 + (size_t)N * 128;
  float* x3 = x2 + (size_t)N * 256;
  float* pooled = x3 + (size_t)N * 512;
  float* norms = pooled + (size_t)NC * 512;

  const int ntiles = (N + 15) / 16;
  const int mlp_blocks = (ntiles + 3) / 4;

  const float* xin = x0;
  float* xouts[3] = {x1, x2, x3};
  int cin = 64;
  for (int l = 0; l < 3; ++l) {
    float* xout = xouts[l];
    glp_mlp_kernel<<<mlp_blocks, 128, 0, stream>>>(
        xin, W1[l], B1[l], G[l], BE[l], W2[l], B2[l], xout, cin, ntiles);

    const int c4sh = (cin == 64) ? 4 : (cin == 128 ? 5 : 6);  // log2(C/4)
    long long sc_total = ((long long)E) << c4sh;
    scatter_max_kernel<<<(int)((sc_total + 255) / 256), 256, 0, stream>>>(
        xout, src, dst, E, cin, c4sh);

    long long fx_total = (long long)N * cin * 2;
    fixup_neginf_kernel<<<(int)((fx_total + 255) / 256), 256, 0, stream>>>(
        xout, fx_total);

    xin = xout;
    cin *= 2;
  }

  const long long pn = (long long)NC * 512;
  init_neginf_kernel<<<(int)((pn + 255) / 256), 256, 0, stream>>>(pooled, pn);
  const long long pt = (long long)N * 128;
  pool_max_kernel<<<(int)((pt + 255) / 256), 256, 0, stream>>>(x3, cluster,
                                                               pooled, N);
  fixup_neginf_kernel<<<(int)((pn + 255) / 256), 256, 0, stream>>>(pooled, pn);
  col_norm_kernel<<<512, 256, 0, stream>>>(pooled, norms, NC);
  normalize_kernel<<<(int)((pn + 255) / 256), 256, 0, stream>>>(
      pooled, norms, (float*)d_out, pn);
}